// AttentionCropper_76605036691482
// MI455X (gfx1250) — compile-verified
//
#include <hip/hip_runtime.h>
#include <stdint.h>

// Problem constants (from reference): B=32, C=3, H=W=448, Hp=Wp=14, TARGET=224.
#define TARGET 224
#define HP 14
#define WP 14
#define IMG 448
#define SCALE 32            // IMG / HP == 448/14, exact -> integer bbox coords
#define ROWS_PER_BLOCK 8    // one output row per wave, 8 waves per block

// ---------------------------------------------------------------------------
// Kernel 1: per-sample bbox from thresholded attention map.
// One wave32 per sample. Emits {y0, size_y, x0, size_x} ints into d_ws.
// All coords are multiples of 32 -> 128B-aligned source spans (b128 legal).
// ---------------------------------------------------------------------------
__global__ __launch_bounds__(32) void attn_bbox_kernel(
    const float* __restrict__ attn, int* __restrict__ bbox)
{
    const int b    = blockIdx.x;
    const int lane = threadIdx.x;
    const float* am = attn + b * (HP * WP);

    // Wave-wide max of the 196 values.
    float vmax = -3.0e38f;
    for (int i = lane; i < HP * WP; i += 32)
        vmax = fmaxf(vmax, am[i]);
    for (int off = 16; off >= 1; off >>= 1)
        vmax = fmaxf(vmax, __shfl_xor(vmax, off, 32));
    const float thresh = vmax * 0.5f;

    // 14-bit row/col "any above threshold" masks.
    unsigned rowm = 0u, colm = 0u;
    for (int i = lane; i < HP * WP; i += 32) {
        if (am[i] > thresh) {
            rowm |= 1u << (i / WP);
            colm |= 1u << (i % WP);
        }
    }
    for (int off = 16; off >= 1; off >>= 1) {
        rowm |= __shfl_xor(rowm, off, 32);
        colm |= __shfl_xor(colm, off, 32);
    }

    if (lane == 0) {
        int y0, y1, x0, x1;
        if (rowm == 0u || colm == 0u) {        // empty mask -> full image
            y0 = 0; y1 = IMG; x0 = 0; x1 = IMG;
        } else {
            const int rmin = __builtin_ctz(rowm);
            const int rmax = 31 - __builtin_clz(rowm);
            const int cmin = __builtin_ctz(colm);
            const int cmax = 31 - __builtin_clz(colm);
            y0 = rmin * SCALE; y1 = (rmax + 1) * SCALE;   // floor(r*32) exact
            x0 = cmin * SCALE; x1 = (cmax + 1) * SCALE;
        }
        bbox[b * 4 + 0] = y0;
        bbox[b * 4 + 1] = y1 - y0;   // size_y
        bbox[b * 4 + 2] = x0;
        bbox[b * 4 + 3] = x1 - x0;   // size_x
    }
}

// ---------------------------------------------------------------------------
// Kernel 2: bilinear crop-resize (align_corners=False), CDNA5 async-LDS path.
// Grid: B * C * (TARGET/ROWS_PER_BLOCK) blocks of 256 threads (8 wave32).
// Each wave owns one output row: async-stage its two source rows [x0,x1)
// into LDS (global_load_async_to_lds_b128 + b32 tail), s_wait_asynccnt,
// then gather the 4 bilinear taps from LDS and write coalesced output.
// ---------------------------------------------------------------------------
__global__ __launch_bounds__(256) void attn_crop_resize_kernel(
    const float* __restrict__ images, const int* __restrict__ bbox,
    float* __restrict__ out)
{
    __shared__ float lds[ROWS_PER_BLOCK * 2 * IMG];   // 28672 B of 320 KB

    const int lane   = threadIdx.x & 31;
    const int waveId = threadIdx.x >> 5;

    const int tilesY = TARGET / ROWS_PER_BLOCK;       // 28
    const int tile = blockIdx.x % tilesY;
    const int c    = (blockIdx.x / tilesY) % 3;
    const int b    =  blockIdx.x / (tilesY * 3);

    const int oy = tile * ROWS_PER_BLOCK + waveId;

    const int y0 = bbox[b * 4 + 0];
    const int sy = bbox[b * 4 + 1];
    const int x0 = bbox[b * 4 + 2];
    const int sx = bbox[b * 4 + 3];

    // y source coords, exact reference math: src=(oy+0.5)*(size/T)-0.5, clip.
    const float fsy = (float)sy;
    float srcy = ((float)oy + 0.5f) * (fsy / (float)TARGET) - 0.5f;
    srcy = fminf(fmaxf(srcy, 0.0f), fsy - 1.0f);
    const int   iy0 = (int)floorf(srcy);
    const int   iy1 = min(iy0 + 1, sy - 1);
    const float fy  = srcy - (float)iy0;

    const size_t plane = (size_t)(b * 3 + c) * (size_t)(IMG * IMG);
    const float* rowl = images + plane + (size_t)(y0 + iy0) * IMG + x0;
    const float* rowh = images + plane + (size_t)(y0 + iy1) * IMG + x0;

    float* ldsl = &lds[waveId * (2 * IMG)];
    float* ldsh = ldsl + IMG;

    // ---- async global -> LDS staging of both source rows ----
    // x0 is a multiple of 32 floats and image row pitch is 1792 B, so every
    // (rowbase + 4*idx) with idx%4==0 is 16B aligned: b128 is legal.
    const int n128 = sx >> 7;                       // 128-float chunks
    for (int k = 0; k < n128; ++k) {
        const int idx = (k << 7) + (lane << 2);
        uint64_t ga; uint32_t la;
        ga = (uint64_t)(uintptr_t)(rowl + idx);
        la = (uint32_t)(uintptr_t)(ldsl + idx);     // low 32b of flat = LDS addr
        asm volatile("global_load_async_to_lds_b128 %0, %1, off"
                     :: "v"(la), "v"(ga) : "memory");
        ga = (uint64_t)(uintptr_t)(rowh + idx);
        la = (uint32_t)(uintptr_t)(ldsh + idx);
        asm volatile("global_load_async_to_lds_b128 %0, %1, off"
                     :: "v"(la), "v"(ga) : "memory");
    }
    const int done = n128 << 7;                     // sx multiple of 32: b32 tail
    const int n32  = (sx - done) >> 5;
    for (int k = 0; k < n32; ++k) {
        const int idx = done + (k << 5) + lane;
        uint64_t ga; uint32_t la;
        ga = (uint64_t)(uintptr_t)(rowl + idx);
        la = (uint32_t)(uintptr_t)(ldsl + idx);
        asm volatile("global_load_async_to_lds_b32 %0, %1, off"
                     :: "v"(la), "v"(ga) : "memory");
        ga = (uint64_t)(uintptr_t)(rowh + idx);
        la = (uint32_t)(uintptr_t)(ldsh + idx);
        asm volatile("global_load_async_to_lds_b32 %0, %1, off"
                     :: "v"(la), "v"(ga) : "memory");
    }
    asm volatile("s_wait_asynccnt 0" ::: "memory"); // ASYNCcnt==0: LDS data ready
    // Only the issuing wave reads its slice -> no cross-wave barrier needed.

    // ---- bilinear resample out of LDS ----
    const float fsx  = (float)sx;
    const float omfy = 1.0f - fy;
    float* orow = out + ((size_t)(b * 3 + c) * TARGET + (size_t)oy) * TARGET;

    #pragma unroll
    for (int j = 0; j < TARGET / 32; ++j) {         // 7 pixels per lane
        const int ox = (j << 5) + lane;
        float srcx = ((float)ox + 0.5f) * (fsx / (float)TARGET) - 0.5f;
        srcx = fminf(fmaxf(srcx, 0.0f), fsx - 1.0f);
        const int   ix0 = (int)floorf(srcx);
        const int   ix1 = min(ix0 + 1, sx - 1);
        const float fx  = srcx - (float)ix0;

        const float tl = ldsl[ix0], tr = ldsl[ix1];
        const float bl = ldsh[ix0], br = ldsh[ix1];
        const float top = tl * (1.0f - fx) + tr * fx;
        const float bot = bl * (1.0f - fx) + br * fx;
        orow[ox] = top * omfy + bot * fy;           // coalesced 128B store
    }
}

// ---------------------------------------------------------------------------
// Launcher
// ---------------------------------------------------------------------------
extern "C" void kernel_launch(void* const* d_in, const int* in_sizes, int n_in,
                              void* d_out, int out_size, void* d_ws, size_t ws_size,
                              hipStream_t stream) {
    const float* attn   = (const float*)d_in[0];   // [32,14,14] f32
    const float* images = (const float*)d_in[1];   // [32,3,448,448] f32
    float*       out    = (float*)d_out;           // [32,3,224,224] f32
    int*         bbox   = (int*)d_ws;              // 32 * int4

    attn_bbox_kernel<<<32, 32, 0, stream>>>(attn, bbox);

    const int blocks = 32 * 3 * (TARGET / ROWS_PER_BLOCK);   // 2688
    attn_crop_resize_kernel<<<blocks, 256, 0, stream>>>(images, bbox, out);
}